// Raindrop_v2_24086176596194
// MI455X (gfx1250) — compile-verified
//
#include <hip/hip_runtime.h>
#include <hip/hip_bf16.h>

typedef float v2f __attribute__((ext_vector_type(2)));
typedef float v8f __attribute__((ext_vector_type(8)));

// ---------- helpers ----------
__device__ __forceinline__ unsigned f2o(float f) {
    unsigned b = __float_as_uint(f);
    return (b & 0x80000000u) ? ~b : (b | 0x80000000u);
}
__device__ __forceinline__ float o2f(unsigned u) {
    unsigned b = (u & 0x80000000u) ? (u & 0x7FFFFFFFu) : ~u;
    return __uint_as_float(b);
}

__global__ void k_zero_u32(unsigned* __restrict__ p, int n) {
    int i = blockIdx.x * blockDim.x + threadIdx.x;
    if (i < n) p[i] = 0u;
}

// ---------- GEMM1 fused: beta_node[d,l] = (1/32) * sum_c (x@W_inc + b_inc)[d, l*32+c] * coef
// coef = mw[d,c] for c<16, p_t[l,c-16] for c>=16.
// One wave computes a 16(M) x 64(N) strip: A fragment reused across 4 WMMA tiles.
__global__ __launch_bounds__(32) void k_gemm_beta(
    const float* __restrict__ x, const float* __restrict__ Winc,
    const float* __restrict__ binc, const float* __restrict__ mw,
    const float* __restrict__ pt, float* __restrict__ beta)
{
    const int lane = threadIdx.x;
    const int n0 = blockIdx.x * 64;        // 2048 / 64 = 32 strips
    const int m0 = blockIdx.y * 16;
    const int colbase = n0 + (lane & 15);
    const int khalf = (lane >> 4) << 1;    // 0 or 2
    const int arow = m0 + (lane & 15);

    v8f acc[4] = {{}, {}, {}, {}};
    for (int k0 = 0; k0 < 256; k0 += 4) {
        const int kb = k0 + khalf;
        v2f a;
        a.x = x[arow * 256 + kb];
        a.y = x[arow * 256 + kb + 1];
        const float* B0 = Winc + kb * 2048 + colbase;
        const float* B1 = Winc + (kb + 1) * 2048 + colbase;
        v2f b0 = {B0[0],  B1[0]};
        v2f b1 = {B0[16], B1[16]};
        v2f b2 = {B0[32], B1[32]};
        v2f b3 = {B0[48], B1[48]};
        acc[0] = __builtin_amdgcn_wmma_f32_16x16x4_f32(false, a, false, b0, (short)0, acc[0], false, false);
        acc[1] = __builtin_amdgcn_wmma_f32_16x16x4_f32(false, a, false, b1, (short)0, acc[1], false, false);
        acc[2] = __builtin_amdgcn_wmma_f32_16x16x4_f32(false, a, false, b2, (short)0, acc[2], false, false);
        acc[3] = __builtin_amdgcn_wmma_f32_16x16x4_f32(false, a, false, b3, (short)0, acc[3], false, false);
    }

    // Epilogue: tiles t=0,2 are mw-coefficient halves (c in 0..15); t=1,3 are p_t halves.
    #pragma unroll
    for (int t = 0; t < 4; ++t) {
        const int col = colbase + 16 * t;
        const int l = col >> 5;
        const int c = col & 31;
        const float bias = binc[col];
        const bool is_pt = (t & 1);                 // wave-uniform per tile
        const float coef_u = is_pt ? pt[l * 16 + (c - 16)] : 0.0f;
        #pragma unroll
        for (int i = 0; i < 8; ++i) {
            const int row = m0 + i + ((lane >> 4) << 3);
            const float h = acc[t][i] + bias;
            const float coef = is_pt ? coef_u : mw[row * 16 + c];
            float s = h * coef * (1.0f / 32.0f);
            s += __shfl_xor(s, 1);
            s += __shfl_xor(s, 2);
            s += __shfl_xor(s, 4);
            s += __shfl_xor(s, 8);
            if ((lane & 15) == 0) atomicAdd(&beta[row * 64 + l], s);
        }
    }
}

// ---------- GEMM2: V = relu(x @ W_value + b_value), (N x 256); 16x64 strip per wave ----------
__global__ __launch_bounds__(32) void k_gemm_value(
    const float* __restrict__ x, const float* __restrict__ Wv,
    const float* __restrict__ bv, float* __restrict__ V)
{
    const int lane = threadIdx.x;
    const int n0 = blockIdx.x * 64;        // 256 / 64 = 4 strips
    const int m0 = blockIdx.y * 16;
    const int colbase = n0 + (lane & 15);
    const int khalf = (lane >> 4) << 1;
    const int arow = m0 + (lane & 15);

    v8f acc[4] = {{}, {}, {}, {}};
    for (int k0 = 0; k0 < 256; k0 += 4) {
        const int kb = k0 + khalf;
        v2f a;
        a.x = x[arow * 256 + kb];
        a.y = x[arow * 256 + kb + 1];
        const float* B0 = Wv + kb * 256 + colbase;
        const float* B1 = Wv + (kb + 1) * 256 + colbase;
        v2f b0 = {B0[0],  B1[0]};
        v2f b1 = {B0[16], B1[16]};
        v2f b2 = {B0[32], B1[32]};
        v2f b3 = {B0[48], B1[48]};
        acc[0] = __builtin_amdgcn_wmma_f32_16x16x4_f32(false, a, false, b0, (short)0, acc[0], false, false);
        acc[1] = __builtin_amdgcn_wmma_f32_16x16x4_f32(false, a, false, b1, (short)0, acc[1], false, false);
        acc[2] = __builtin_amdgcn_wmma_f32_16x16x4_f32(false, a, false, b2, (short)0, acc[2], false, false);
        acc[3] = __builtin_amdgcn_wmma_f32_16x16x4_f32(false, a, false, b3, (short)0, acc[3], false, false);
    }
    #pragma unroll
    for (int t = 0; t < 4; ++t) {
        const int col = colbase + 16 * t;
        const float bias = bv[col];
        #pragma unroll
        for (int i = 0; i < 8; ++i) {
            const int row = m0 + i + ((lane >> 4) << 3);
            V[row * 256 + col] = fmaxf(acc[t][i] + bias, 0.0f);
        }
    }
}

// ---------- per-node mean of beta ----------
__global__ void k_beta_mean(const float* __restrict__ beta, float* __restrict__ bmean, int N) {
    int n = blockIdx.x * blockDim.x + threadIdx.x;
    if (n >= N) return;
    float s = 0.f;
    #pragma unroll
    for (int l = 0; l < 64; ++l) s += beta[n * 64 + l];
    bmean[n] = s * (1.0f / 64.0f);
}

// ---------- per-edge key: ordered-uint of ew[e] * bmean[dst[e]] ----------
__global__ void k_keys(const int* __restrict__ eidx, const float* __restrict__ ew,
                       const float* __restrict__ bmean, unsigned* __restrict__ keys, int E) {
    int e = blockIdx.x * blockDim.x + threadIdx.x;
    if (e >= E) return;
    int d = eidx[E + e];
    keys[e] = f2o(ew[e] * bmean[d]);
}

// ---------- radix select: K-th largest key (single block) ----------
__global__ __launch_bounds__(256) void k_radix_select(const unsigned* __restrict__ keys,
                                                      int E, int K, unsigned* __restrict__ scal) {
    __shared__ unsigned hist[256];
    __shared__ unsigned s_prefix;
    __shared__ unsigned s_rem;
    const int tid = threadIdx.x;
    if (tid == 0) { s_prefix = 0u; s_rem = (unsigned)K; }
    __syncthreads();
    #pragma unroll
    for (int shift = 24; shift >= 0; shift -= 8) {
        hist[tid] = 0u;
        __syncthreads();
        const unsigned mask_known = (shift == 24) ? 0u : (0xFFFFFFFFu << (shift + 8));
        const unsigned pref = s_prefix;
        for (int e = tid; e < E; e += 256) {
            unsigned u = keys[e];
            if ((u & mask_known) == pref) atomicAdd(&hist[(u >> shift) & 255u], 1u);
        }
        __syncthreads();
        if (tid == 0) {
            unsigned rem = s_rem, cum = 0u;
            int b = 255;
            for (; b > 0; --b) {
                if (cum + hist[b] >= rem) break;
                cum += hist[b];
            }
            s_prefix |= ((unsigned)b) << shift;
            s_rem = rem - cum;
        }
        __syncthreads();
    }
    if (tid == 0) { scal[0] = s_prefix; scal[1] = s_rem; }  // threshold, #ties needed
}

// ---------- selection marking ----------
__global__ void k_mark_gt(const unsigned* __restrict__ keys, const unsigned* __restrict__ scal,
                          unsigned* __restrict__ sel, int E) {
    int e = blockIdx.x * blockDim.x + threadIdx.x;
    if (e >= E) return;
    sel[e] = (keys[e] > scal[0]) ? 1u : 0u;
}

__global__ __launch_bounds__(32) void k_tie_cnt(const unsigned* __restrict__ keys,
                                                const unsigned* __restrict__ scal,
                                                unsigned* __restrict__ tiecnt, int E) {
    int e = blockIdx.x * 32 + threadIdx.x;
    bool t = (e < E) && (keys[e] == scal[0]);
    unsigned long long m = __ballot(t);
    if (threadIdx.x == 0) tiecnt[blockIdx.x] = (unsigned)__popcll(m & 0xFFFFFFFFull);
}

__global__ void k_tie_prefix(const unsigned* __restrict__ tiecnt,
                             unsigned* __restrict__ tiepre, int chunks) {
    unsigned run = 0;
    for (int i = 0; i < chunks; ++i) { tiepre[i] = run; run += tiecnt[i]; }
}

__global__ __launch_bounds__(32) void k_tie_mark(const unsigned* __restrict__ keys,
                                                 const unsigned* __restrict__ scal,
                                                 const unsigned* __restrict__ tiepre,
                                                 unsigned* __restrict__ sel, int E) {
    int e = blockIdx.x * 32 + threadIdx.x;
    bool t = (e < E) && (keys[e] == scal[0]);
    unsigned long long m = __ballot(t);
    unsigned lanemask = (threadIdx.x == 0) ? 0u : (0xFFFFFFFFu >> (32 - threadIdx.x));
    unsigned rank = (unsigned)__popcll(m & (unsigned long long)lanemask);
    if (t && (tiepre[blockIdx.x] + rank) < scal[1]) sel[e] = 1u;
}

// ---------- segment softmax (per src node, per l) + scatter ----------
__global__ void k_segmax(const int* __restrict__ eidx, const float* __restrict__ ew,
                         const float* __restrict__ beta, const unsigned* __restrict__ sel,
                         unsigned* __restrict__ smax, int E) {
    int e = blockIdx.x * blockDim.x + threadIdx.x;
    if (e >= E || !sel[e]) return;
    int s = eidx[e], d = eidx[E + e];
    float w = ew[e];
    #pragma unroll 4
    for (int l = 0; l < 64; ++l) {
        float g = beta[d * 64 + l] * w;
        atomicMax(&smax[s * 64 + l], f2o(g));
    }
}

__global__ void k_segsum(const int* __restrict__ eidx, const float* __restrict__ ew,
                         const float* __restrict__ beta, const unsigned* __restrict__ sel,
                         const unsigned* __restrict__ smax, float* __restrict__ ssum, int E) {
    int e = blockIdx.x * blockDim.x + threadIdx.x;
    if (e >= E || !sel[e]) return;
    int s = eidx[e], d = eidx[E + e];
    float w = ew[e];
    #pragma unroll 4
    for (int l = 0; l < 64; ++l) {
        float g = beta[d * 64 + l] * w;
        float m = o2f(smax[s * 64 + l]);
        atomicAdd(&ssum[s * 64 + l], expf(g - m));
    }
}

__global__ void k_scatter(const int* __restrict__ eidx, const float* __restrict__ ew,
                          const float* __restrict__ beta, const unsigned* __restrict__ sel,
                          const unsigned* __restrict__ smax, const float* __restrict__ ssum,
                          const float* __restrict__ V, float* __restrict__ out, int E) {
    int e = blockIdx.x * blockDim.x + threadIdx.x;
    if (e >= E || !sel[e]) return;
    int s = eidx[e], d = eidx[E + e];
    float w = ew[e];
    #pragma unroll 2
    for (int l = 0; l < 64; ++l) {
        float g = beta[d * 64 + l] * w;
        float m = o2f(smax[s * 64 + l]);
        float wgt = expf(g - m) / (ssum[s * 64 + l] + 1e-16f);
        #pragma unroll
        for (int o = 0; o < 4; ++o) {
            int ch = l * 4 + o;
            atomicAdd(&out[s * 256 + ch], V[d * 256 + ch] * wgt);
        }
    }
}

// ---------- host ----------
extern "C" void kernel_launch(void* const* d_in, const int* in_sizes, int n_in,
                              void* d_out, int out_size, void* d_ws, size_t ws_size,
                              hipStream_t stream) {
    const float* x     = (const float*)d_in[0];
    const float* p_t   = (const float*)d_in[1];
    const int*   eidx  = (const int*)d_in[2];
    const float* ew    = (const float*)d_in[3];
    const float* Wv    = (const float*)d_in[4];
    const float* bv    = (const float*)d_in[5];
    const float* Winc  = (const float*)d_in[6];
    const float* binc  = (const float*)d_in[7];
    const float* mw    = (const float*)d_in[8];

    const int N = in_sizes[0] / 256;          // 4000
    const int E = in_sizes[3];                // 64000
    const int K = E / 2;                      // 32000
    const int chunks = (E + 31) / 32;

    // workspace carve-out (all 4-byte elements)
    char* ws = (char*)d_ws;
    float*    beta   = (float*)ws;                 ws += (size_t)N * 64 * 4;
    float*    V      = (float*)ws;                 ws += (size_t)N * 256 * 4;
    float*    bmean  = (float*)ws;                 ws += (size_t)N * 4;
    unsigned* keys   = (unsigned*)ws;              ws += (size_t)E * 4;
    unsigned* sel    = (unsigned*)ws;              ws += (size_t)E * 4;
    unsigned* tiecnt = (unsigned*)ws;              ws += (size_t)chunks * 4;
    unsigned* tiepre = (unsigned*)ws;              ws += (size_t)chunks * 4;
    unsigned* scal   = (unsigned*)ws;              ws += 8 * 4;
    unsigned* smax   = (unsigned*)ws;              ws += (size_t)N * 64 * 4;
    float*    ssum   = (float*)ws;                 ws += (size_t)N * 64 * 4;

    // zero accumulators (+ output)
    k_zero_u32<<<(N * 64 + 255) / 256, 256, 0, stream>>>((unsigned*)beta, N * 64);
    k_zero_u32<<<(N * 64 + 255) / 256, 256, 0, stream>>>(smax, N * 64);
    k_zero_u32<<<(N * 64 + 255) / 256, 256, 0, stream>>>((unsigned*)ssum, N * 64);
    k_zero_u32<<<(out_size + 255) / 256, 256, 0, stream>>>((unsigned*)d_out, out_size);

    // per-node GEMMs (WMMA f32 16x16x4, 16x64 strip per wave)
    k_gemm_beta<<<dim3(2048 / 64, N / 16), 32, 0, stream>>>(x, Winc, binc, mw, p_t, beta);
    k_gemm_value<<<dim3(256 / 64, N / 16), 32, 0, stream>>>(x, Wv, bv, V);

    // edge scoring + top-K selection
    k_beta_mean<<<(N + 255) / 256, 256, 0, stream>>>(beta, bmean, N);
    k_keys<<<(E + 255) / 256, 256, 0, stream>>>(eidx, ew, bmean, keys, E);
    k_radix_select<<<1, 256, 0, stream>>>(keys, E, K, scal);
    k_mark_gt<<<(E + 255) / 256, 256, 0, stream>>>(keys, scal, sel, E);
    k_tie_cnt<<<chunks, 32, 0, stream>>>(keys, scal, tiecnt, E);
    k_tie_prefix<<<1, 1, 0, stream>>>(tiecnt, tiepre, chunks);
    k_tie_mark<<<chunks, 32, 0, stream>>>(keys, scal, tiepre, sel, E);

    // segment softmax + scatter
    k_segmax<<<(E + 255) / 256, 256, 0, stream>>>(eidx, ew, beta, sel, smax, E);
    k_segsum<<<(E + 255) / 256, 256, 0, stream>>>(eidx, ew, beta, sel, smax, ssum, E);
    k_scatter<<<(E + 255) / 256, 256, 0, stream>>>(eidx, ew, beta, sel, smax, ssum, V, (float*)d_out, E);
}